// Model_24945170055652
// MI455X (gfx1250) — compile-verified
//
#include <hip/hip_runtime.h>

#define N_NODES    1000000
#define N_EDGES    32000000
#define NUM_GRAPHS 1024

#define EDGE_BLOCK 256
#define TILE       1024           // edges staged per block-tile (256 thr * 4 edges)
#define EDGE_GRID  2048
#define NODES_PER_THREAD 16

// ---- CDNA5 async global->LDS helpers (ASYNCcnt path) -----------------------

// Low 32 bits of a flat pointer into LDS == wave-relative LDS byte address.
__device__ __forceinline__ unsigned lds_off_of(const void* p) {
    return (unsigned)(unsigned long long)p;
}

// GVS form: lds[vdst] <= mem[ saddr(SGPR64) + vaddr(VGPR32) ], 16 bytes/lane.
__device__ __forceinline__ void async_copy_b128(unsigned lds_off, unsigned gbyte_off,
                                                const int* base) {
    asm volatile("global_load_async_to_lds_b128 %0, %1, %2"
                 :: "v"(lds_off), "v"(gbyte_off), "s"(base)
                 : "memory");
}

__device__ __forceinline__ void wait_async0() {
    asm volatile("s_wait_asynccnt 0" ::: "memory");
}

// Native FP32 add atomic at device scope (L2 atomic units), no return ->
// STOREcnt-tracked, s_endpgm drains implicitly. Guarantees we never fall
// back to a cmpswap retry loop.
__device__ __forceinline__ void atomic_add_f32_dev(float* base, int idx, float val) {
    asm volatile("global_atomic_add_f32 %0, %1, %2 scope:SCOPE_DEV"
                 :: "v"((unsigned)idx * 4u), "v"(val), "s"(base)
                 : "memory");
}

// ---- Kernel 1: zero scratch ------------------------------------------------

__global__ void zero_kernel(float* __restrict__ p, int n) {
    int i = blockIdx.x * blockDim.x + threadIdx.x;
    if (i < n) p[i] = 0.0f;
}

// ---- Kernel 2: edge scatter-add  agg[dst] += x[src] ------------------------
// Double-buffered async staging of the streaming edge-index data into LDS,
// overlapping next tile's stream with current tile's gathers + L2 atomics.

__global__ void __launch_bounds__(EDGE_BLOCK)
edge_scatter_kernel(const int* __restrict__ src, const int* __restrict__ dst,
                    const float* __restrict__ x, float* __restrict__ agg,
                    int numTiles) {
    __shared__ int s_src[2][TILE];
    __shared__ int s_dst[2][TILE];
    const int tid = threadIdx.x;

    auto stage = [&](int buf, int tile) {
        unsigned gb = (unsigned)((tile * TILE + tid * 4) * 4); // byte offset
        async_copy_b128(lds_off_of(&s_src[buf][tid * 4]), gb, src);
        async_copy_b128(lds_off_of(&s_dst[buf][tid * 4]), gb, dst);
    };

    int tile = blockIdx.x;
    int buf  = 0;
    if (tile < numTiles) stage(0, tile);

    for (; tile < numTiles; tile += gridDim.x) {
        wait_async0();        // this wave's staged loads for `buf` are in LDS
        __syncthreads();      // all waves staged + everyone done with buf^1
        int next = tile + gridDim.x;
        if (next < numTiles) stage(buf ^ 1, next);   // overlap with compute

        // Wide LDS reads (ds_load_b128 x2), then 4 gathers + 4 L2 atomics.
        const int4 sv = *reinterpret_cast<const int4*>(&s_src[buf][tid * 4]);
        const int4 dv = *reinterpret_cast<const int4*>(&s_dst[buf][tid * 4]);
        atomic_add_f32_dev(agg, dv.x, x[sv.x]);
        atomic_add_f32_dev(agg, dv.y, x[sv.y]);
        atomic_add_f32_dev(agg, dv.z, x[sv.z]);
        atomic_add_f32_dev(agg, dv.w, x[sv.w]);

        buf ^= 1;
    }
}

// ---- Kernel 3: ReLU + sorted segmented pool (run-length -> few atomics) ----

__global__ void pool_kernel(const float* __restrict__ agg,
                            const int* __restrict__ batch,
                            float* __restrict__ sums,
                            float* __restrict__ counts) {
    int t = blockIdx.x * blockDim.x + threadIdx.x;
    int start = t * NODES_PER_THREAD;
    if (start >= N_NODES) return;
    // N_NODES is a multiple of 4; start is 16-aligned -> int4/float4 safe.

    int   cur = batch[start];
    float s = 0.0f, c = 0.0f;

    #pragma unroll
    for (int q = 0; q < NODES_PER_THREAD / 4; ++q) {
        int n = start + q * 4;
        if (n >= N_NODES) break;
        const int4   bv = *reinterpret_cast<const int4*>(&batch[n]);
        const float4 av = *reinterpret_cast<const float4*>(&agg[n]);
        const int   bb[4] = { bv.x, bv.y, bv.z, bv.w };
        const float aa[4] = { av.x, av.y, av.z, av.w };
        #pragma unroll
        for (int k = 0; k < 4; ++k) {
            float h = aa[k] > 0.0f ? aa[k] : 0.0f;   // relu
            if (bb[k] != cur) {
                atomic_add_f32_dev(sums,   cur, s);
                atomic_add_f32_dev(counts, cur, c);
                cur = bb[k]; s = 0.0f; c = 0.0f;
            }
            s += h; c += 1.0f;
        }
    }
    atomic_add_f32_dev(sums,   cur, s);
    atomic_add_f32_dev(counts, cur, c);
}

// ---- Kernel 4: pooled = sums/max(counts,1);  out = pooled*W + b ------------

__global__ void finalize_kernel(const float* __restrict__ sums,
                                const float* __restrict__ counts,
                                const float* __restrict__ W,
                                const float* __restrict__ b,
                                float* __restrict__ out) {
    int g = blockIdx.x * blockDim.x + threadIdx.x;
    if (g < NUM_GRAPHS) {
        float c = counts[g];
        c = c > 1.0f ? c : 1.0f;
        out[g] = (sums[g] / c) * W[0] + b[0];
    }
}

// ---- Host-side launcher ----------------------------------------------------

extern "C" void kernel_launch(void* const* d_in, const int* in_sizes, int n_in,
                              void* d_out, int out_size, void* d_ws, size_t ws_size,
                              hipStream_t stream) {
    const float* x          = (const float*)d_in[0];
    const float* W          = (const float*)d_in[1];
    const float* b          = (const float*)d_in[2];
    const int*   edge_index = (const int*)d_in[3];   // [2, N_EDGES] flat
    const int*   batch      = (const int*)d_in[4];
    float*       out        = (float*)d_out;

    float* agg    = (float*)d_ws;            // N_NODES floats
    float* sums   = agg + N_NODES;           // NUM_GRAPHS floats
    float* counts = sums + NUM_GRAPHS;       // NUM_GRAPHS floats

    // 1) zero agg + sums + counts
    {
        int n = N_NODES + 2 * NUM_GRAPHS;
        zero_kernel<<<(n + 255) / 256, 256, 0, stream>>>(agg, n);
    }
    // 2) edge scatter-add (async-LDS double-buffered streaming)
    {
        const int* src = edge_index;
        const int* dst = edge_index + N_EDGES;
        int numTiles = N_EDGES / TILE;       // 31250, exact
        edge_scatter_kernel<<<EDGE_GRID, EDGE_BLOCK, 0, stream>>>(
            src, dst, x, agg, numTiles);
    }
    // 3) relu + per-graph sum/count (batch is sorted -> run-length atomics)
    {
        int nThreads = (N_NODES + NODES_PER_THREAD - 1) / NODES_PER_THREAD;
        pool_kernel<<<(nThreads + 255) / 256, 256, 0, stream>>>(
            agg, batch, sums, counts);
    }
    // 4) mean + linear
    finalize_kernel<<<(NUM_GRAPHS + 255) / 256, 256, 0, stream>>>(
        sums, counts, W, b, out);
}